// TemporalEdgeClassifier_87711822119150
// MI455X (gfx1250) — compile-verified
//
#include <hip/hip_runtime.h>

#define NN 100000
#define EE 600000
#define HD 128
#define EDIM 32

typedef __attribute__((ext_vector_type(2))) float v2f;
typedef __attribute__((ext_vector_type(8))) float v8f;

__device__ __forceinline__ v8f wmma4(v2f a, v2f b, v8f c) {
    // D(16x16,f32) = A(16x4,f32) x B(4x16,f32) + C
    return __builtin_amdgcn_wmma_f32_16x16x4_f32(
        /*neg_a=*/false, a, /*neg_b=*/false, b,
        /*c_mod=*/(short)0, c, /*reuse_a=*/false, /*reuse_b=*/false);
}

__device__ __forceinline__ float sigmoidf_(float x) {
    return 1.0f / (1.0f + __expf(-x));
}

// ---------------------------------------------------------------------------
// Fused GRU cell: h' = (1-z)*n + z*h for a 16-node tile per block.
// 8 waves per block; wave j owns output columns [j*16, j*16+16).
// ---------------------------------------------------------------------------
__global__ void __launch_bounds__(256) gru_kernel(
    const float* __restrict__ x, const float* __restrict__ h,
    const float* __restrict__ Wih, const float* __restrict__ Whh,
    const float* __restrict__ bih, const float* __restrict__ bhh,
    float* __restrict__ hout)
{
    __shared__ __align__(16) float sX[16 * 132];
    __shared__ __align__(16) float sH[16 * 132];
    const int tid = threadIdx.x;
    const int nbase = blockIdx.x * 16;

    for (int i = tid; i < 16 * 32; i += 256) {
        int row = i >> 5, c4 = (i & 31) * 4;
        *(float4*)(sX + row * 132 + c4) =
            *(const float4*)(x + (size_t)(nbase + row) * HD + c4);
        *(float4*)(sH + row * 132 + c4) =
            *(const float4*)(h + (size_t)(nbase + row) * HD + c4);
    }
    __syncthreads();

    const int j = tid >> 5;
    const int lane = tid & 31;
    const int m = lane & 15;
    const int hh = lane >> 4;
    const int n = lane & 15;
    const int col = j * 16 + n;

    const float* wiR = Wih + (size_t)(col) * HD;
    const float* wiZ = Wih + (size_t)(HD + col) * HD;
    const float* wiN = Wih + (size_t)(2 * HD + col) * HD;
    const float* whR = Whh + (size_t)(col) * HD;
    const float* whZ = Whh + (size_t)(HD + col) * HD;
    const float* whN = Whh + (size_t)(2 * HD + col) * HD;

    v8f aR = {}, aZ = {}, aN = {}, gR = {}, gZ = {}, gN = {};
    for (int kt = 0; kt < 32; ++kt) {
        const int k = kt * 4 + 2 * hh;
        v2f ax = { sX[m * 132 + k], sX[m * 132 + k + 1] };
        v2f ah = { sH[m * 132 + k], sH[m * 132 + k + 1] };
        v2f bR = { wiR[k], wiR[k + 1] };
        v2f bZ = { wiZ[k], wiZ[k + 1] };
        v2f bN = { wiN[k], wiN[k + 1] };
        v2f cR = { whR[k], whR[k + 1] };
        v2f cZ = { whZ[k], whZ[k + 1] };
        v2f cN = { whN[k], whN[k + 1] };
        aR = wmma4(ax, bR, aR);
        aZ = wmma4(ax, bZ, aZ);
        aN = wmma4(ax, bN, aN);
        gR = wmma4(ah, cR, gR);
        gZ = wmma4(ah, cZ, gZ);
        gN = wmma4(ah, cN, gN);
    }

    const float biR = bih[col], biZ = bih[HD + col], biN = bih[2 * HD + col];
    const float bhR = bhh[col], bhZ = bhh[HD + col], bhN = bhh[2 * HD + col];
    for (int r = 0; r < 8; ++r) {
        int rowm = r + 8 * hh;
        float ir = aR[r] + biR, iz = aZ[r] + biZ, inn = aN[r] + biN;
        float hr = gR[r] + bhR, hz = gZ[r] + bhZ, hn = gN[r] + bhN;
        float rg = sigmoidf_(ir + hr);
        float zg = sigmoidf_(iz + hz);
        float ng = tanhf(inn + rg * hn);
        float hold = sH[rowm * 132 + col];
        hout[(size_t)(nbase + rowm) * HD + col] = (1.0f - zg) * ng + zg * hold;
    }
}

// ---------------------------------------------------------------------------
// Per-target in-degree counts (float) via atomics.
// ---------------------------------------------------------------------------
__global__ void __launch_bounds__(256) count_kernel(
    const int* __restrict__ dst, float* __restrict__ cnt, int e_total)
{
    int t = blockIdx.x * 256 + threadIdx.x;
    if (t < e_total) atomicAdd(&cnt[dst[t]], 1.0f);
}

__global__ void __launch_bounds__(256) zero_kernel(float4* __restrict__ p, int n4)
{
    int t = blockIdx.x * 256 + threadIdx.x;
    if (t < n4) p[t] = make_float4(0.f, 0.f, 0.f, 0.f);
}

// ---------------------------------------------------------------------------
// Edge scatter: agg[dst] += h[src].  One wave per edge, float4 per lane.
// ---------------------------------------------------------------------------
__global__ void __launch_bounds__(256) scatter_kernel(
    const float* __restrict__ hin, const int* __restrict__ src,
    const int* __restrict__ dst, float* __restrict__ agg)
{
    const int tid = threadIdx.x;
    const int e = blockIdx.x * 8 + (tid >> 5);
    const int lane = tid & 31;
    const int s = src[e];
    const int d = dst[e];
    float4 v = *(const float4*)(hin + (size_t)s * HD + lane * 4);
    float* p = agg + (size_t)d * HD + lane * 4;
    atomicAdd(p + 0, v.x);
    atomicAdd(p + 1, v.y);
    atomicAdd(p + 2, v.z);
    atomicAdd(p + 3, v.w);
}

// ---------------------------------------------------------------------------
// Fused SAGE layer: hout = relu(mean @ Wl.T + bl + hin @ Wr.T).
// Mean normalization folded into LDS staging.  In-place (hout==hin) safe.
// ---------------------------------------------------------------------------
__global__ void __launch_bounds__(256) sage_kernel(
    const float* __restrict__ hin, const float* __restrict__ agg,
    const float* __restrict__ cnt,
    const float* __restrict__ Wl, const float* __restrict__ bl,
    const float* __restrict__ Wr, float* __restrict__ hout)
{
    __shared__ __align__(16) float sM[16 * 132];
    __shared__ __align__(16) float sH[16 * 132];
    const int tid = threadIdx.x;
    const int nbase = blockIdx.x * 16;

    for (int i = tid; i < 16 * 32; i += 256) {
        int row = i >> 5, c4 = (i & 31) * 4;
        float inv = 1.0f / fmaxf(cnt[nbase + row], 1.0f);
        float4 a4 = *(const float4*)(agg + (size_t)(nbase + row) * HD + c4);
        a4.x *= inv; a4.y *= inv; a4.z *= inv; a4.w *= inv;
        *(float4*)(sM + row * 132 + c4) = a4;
        *(float4*)(sH + row * 132 + c4) =
            *(const float4*)(hin + (size_t)(nbase + row) * HD + c4);
    }
    __syncthreads();

    const int j = tid >> 5;
    const int lane = tid & 31;
    const int m = lane & 15;
    const int hh = lane >> 4;
    const int n = lane & 15;
    const int col = j * 16 + n;
    const float* wl = Wl + (size_t)col * HD;
    const float* wr = Wr + (size_t)col * HD;

    v8f aL = {}, aR = {};
    for (int kt = 0; kt < 32; ++kt) {
        const int k = kt * 4 + 2 * hh;
        v2f am = { sM[m * 132 + k], sM[m * 132 + k + 1] };
        v2f ah = { sH[m * 132 + k], sH[m * 132 + k + 1] };
        v2f bL = { wl[k], wl[k + 1] };
        v2f bR = { wr[k], wr[k + 1] };
        aL = wmma4(am, bL, aL);
        aR = wmma4(ah, bR, aR);
    }
    const float bias = bl[col];
    for (int r = 0; r < 8; ++r) {
        int rowm = r + 8 * hh;
        float v = aL[r] + aR[r] + bias;
        hout[(size_t)(nbase + rowm) * HD + col] = fmaxf(v, 0.0f);
    }
}

// ---------------------------------------------------------------------------
// Edge classifier: out = relu([h_s|h_d|e_attr] @ Wc1.T + bc1) @ Wc2.T + bc2.
// 16-edge tile per block; K = 288 (= 72 WMMA steps); 256 hidden cols.
// ---------------------------------------------------------------------------
__global__ void __launch_bounds__(256) edge_cls_kernel(
    const float* __restrict__ hfin, const int* __restrict__ src,
    const int* __restrict__ dst, const float* __restrict__ eattr,
    const float* __restrict__ Wc1, const float* __restrict__ bc1,
    const float* __restrict__ Wc2, const float* __restrict__ bc2,
    float* __restrict__ out)
{
    __shared__ __align__(16) float sA[16 * 292];
    __shared__ __align__(16) float sHid[16 * 260];
    __shared__ __align__(16) float sRed[256];
    const int tid = threadIdx.x;
    const int ebase = blockIdx.x * 16;

    for (int i = tid; i < 16 * 72; i += 256) {
        int row = i / 72, c4 = i % 72;
        int e = ebase + row;
        float4 v;
        if (c4 < 32)
            v = *(const float4*)(hfin + (size_t)src[e] * HD + c4 * 4);
        else if (c4 < 64)
            v = *(const float4*)(hfin + (size_t)dst[e] * HD + (c4 - 32) * 4);
        else
            v = *(const float4*)(eattr + (size_t)e * EDIM + (c4 - 64) * 4);
        *(float4*)(sA + row * 292 + c4 * 4) = v;
    }
    __syncthreads();

    const int j = tid >> 5;
    const int lane = tid & 31;
    const int m = lane & 15;
    const int hh = lane >> 4;
    const int n = lane & 15;

    for (int t2 = 0; t2 < 2; ++t2) {
        const int col = (j + t2 * 8) * 16 + n;
        const float* w = Wc1 + (size_t)col * 288;
        v8f acc = {};
        for (int kt = 0; kt < 72; ++kt) {
            const int k = kt * 4 + 2 * hh;
            v2f a = { sA[m * 292 + k], sA[m * 292 + k + 1] };
            v2f b = { w[k], w[k + 1] };
            acc = wmma4(a, b, acc);
        }
        const float bias = bc1[col];
        for (int r = 0; r < 8; ++r)
            sHid[(r + 8 * hh) * 260 + col] = fmaxf(acc[r] + bias, 0.0f);
    }
    __syncthreads();

    const int e = tid >> 4;
    const int seg = tid & 15;
    float p = 0.f;
    for (int i = 0; i < 16; ++i)
        p += sHid[e * 260 + seg * 16 + i] * Wc2[seg * 16 + i];
    sRed[tid] = p;
    __syncthreads();
    if (seg == 0) {
        float s = bc2[0];
        for (int i = 0; i < 16; ++i) s += sRed[(e << 4) + i];
        out[ebase + e] = s;
    }
}

// ---------------------------------------------------------------------------
extern "C" void kernel_launch(void* const* d_in, const int* in_sizes, int n_in,
                              void* d_out, int out_size, void* d_ws, size_t ws_size,
                              hipStream_t stream)
{
    const float* x    = (const float*)d_in[0];
    const float* h0   = (const float*)d_in[1];
    const int*   eidx = (const int*)d_in[2];
    const float* eatt = (const float*)d_in[3];
    const float* Wih  = (const float*)d_in[4];
    const float* Whh  = (const float*)d_in[5];
    const float* bih  = (const float*)d_in[6];
    const float* bhh  = (const float*)d_in[7];
    const float* Wl[3] = { (const float*)d_in[8],  (const float*)d_in[11], (const float*)d_in[14] };
    const float* blv[3] = { (const float*)d_in[9], (const float*)d_in[12], (const float*)d_in[15] };
    const float* Wr[3] = { (const float*)d_in[10], (const float*)d_in[13], (const float*)d_in[16] };
    const float* Wc1  = (const float*)d_in[17];
    const float* bc1  = (const float*)d_in[18];
    const float* Wc2  = (const float*)d_in[19];
    const float* bc2  = (const float*)d_in[20];

    const int* src = eidx;
    const int* dst = eidx + EE;

    float* out  = (float*)d_out;           // [E] logits
    float* hfin = (float*)d_out + EE;      // [N, H] final node embeddings

    float* hbuf = (float*)d_ws;                 // N*H
    float* agg  = hbuf + (size_t)NN * HD;       // N*H
    float* cnt  = agg + (size_t)NN * HD;        // N

    dim3 blk(256);

    // degree counts (reused by all three layers)
    zero_kernel<<<(NN / 4 + 255) / 256, blk, 0, stream>>>((float4*)cnt, NN / 4);
    count_kernel<<<(EE + 255) / 256, blk, 0, stream>>>(dst, cnt, EE);

    // GRU update
    gru_kernel<<<NN / 16, blk, 0, stream>>>(x, h0, Wih, Whh, bih, bhh, hbuf);

    // three SAGE-mean layers
    for (int L = 0; L < 3; ++L) {
        zero_kernel<<<((NN * HD) / 4 + 255) / 256, blk, 0, stream>>>(
            (float4*)agg, (NN * HD) / 4);
        scatter_kernel<<<EE / 8, blk, 0, stream>>>(hbuf, src, dst, agg);
        float* o = (L == 2) ? hfin : hbuf;
        sage_kernel<<<NN / 16, blk, 0, stream>>>(hbuf, agg, cnt,
                                                 Wl[L], blv[L], Wr[L], o);
    }

    // edge classifier
    edge_cls_kernel<<<EE / 16, blk, 0, stream>>>(hfin, src, dst, eatt,
                                                 Wc1, bc1, Wc2, bc2, out);
}